// AliasFreeActivation_23063974380022
// MI455X (gfx1250) — compile-verified
//
#include <hip/hip_runtime.h>
#include <hip/hip_bf16.h>

// Alias-free activation, fully fused: bias -> up4(24tap) -> lrelu*sqrt2 -> down2(12tap) -> crop10.
// One wave32 per workgroup computes one 16x16 output tile of one (b,c) plane.
// All four separable FIR stages run on the fp32 matrix pipe (V_WMMA_F32_16X16X4_F32).
// Toeplitz tap pairs come from pre-paired float2 LDS LUTs (one ds_load_b64, correct order);
// staging stores are unguarded ds_store_b128 into consumer-transposed LDS buffers.

#define IN_HW    128
#define OUT_HW   236
#define NTILE    15           // ceil(236/16)
#define TILE     16
#define ZSZ      42           // up-grid receptive patch per tile
#define PATCH    17           // input receptive patch per tile
#define SQRT2    1.41421356237309504880f
#define ACT_C1   (0.6f * SQRT2)   // (1+0.2)/2 * sqrt2
#define ACT_C2   (0.4f * SQRT2)   // (1-0.2)/2 * sqrt2

// padded pair-LUT index ranges (OOB taps are built as 0):
#define FU_OFF   56           // up taps: t in [-56, 70]  -> index t+56, len 128
#define FU_LEN   128
#define FD_OFF   32           // down taps: t in [-32, 41] -> index t+32, len 80
#define FD_LEN   80

typedef float v2f __attribute__((ext_vector_type(2)));
typedef float v8f __attribute__((ext_vector_type(8)));

static __device__ __forceinline__ v8f wmma4(v2f a, v2f b, v8f c) {
    // (neg_a, A, neg_b, B, c_mod, C, reuse_a, reuse_b) -> v_wmma_f32_16x16x4_f32
    return __builtin_amdgcn_wmma_f32_16x16x4_f32(false, a, false, b, (short)0, c, false, false);
}

__global__ __launch_bounds__(32)
void afa_fused_kernel(const float* __restrict__ x,
                      const float* __restrict__ bias,
                      const float* __restrict__ upfilt,
                      const float* __restrict__ dnfilt,
                      float* __restrict__ out)
{
    __shared__ v2f  fupair[FU_LEN];  // fupair[i] = { fu[i-FU_OFF], fu[i-FU_OFF-4] }  (taps for k, k+1)
    __shared__ v2f  fdpair[FD_LEN];  // fdpair[i] = { fd[i-FD_OFF], fd[i-FD_OFF-1] }
    __shared__ float xp[32][20];     // biased input patch (17x17 used), zero padded
    __shared__ float ubt[48][32];    // after up-x, TRANSPOSED: [z-col j][input row a]; fully overwritten
    __shared__ float zbt[48][48];    // after up-y+lrelu, TRANSPOSED: [z-col j][z-row p]; fully overwritten
    __shared__ float tbt[16][48];    // after down-x, TRANSPOSED: [out-col mm][z-row p]; fully overwritten

    const int lane = threadIdx.x;    // 0..31 (wave32)
    const int lo   = lane & 15;
    const int hi   = lane >> 4;

    const int plane = blockIdx.y;               // b*256 + c
    const int tile  = blockIdx.x;
    const int q0x = (tile % NTILE) * TILE;
    const int q0y = (tile / NTILE) * TILE;

    // Derived from reference upfirdn padding:
    //   up:   y[n] = sum_i x[i] * fu[n + 10 - 4i]
    //   down: o[m] = sum_n z[n] * fd[2m + 11 - (n - Z0)]  (tile-relative)
    const int Z0x = 2*q0x + 15;
    const int Z0y = 2*q0y + 15;
    const int I0x = (Z0x - 10) >> 2;
    const int I0y = (Z0y - 10) >> 2;
    const int basex = Z0x + 10 - 4*I0x;   // in [20,23]
    const int basey = Z0y + 10 - 4*I0y;

    // ---- build pre-paired tap LUTs (zero outside filter support) ----
    for (int i = lane; i < FU_LEN; i += 32) {
        const int t = i - FU_OFF;
        v2f p;
        p.x = (t     >= 0 && t     < 24) ? upfilt[t]     : 0.f;
        p.y = (t - 4 >= 0 && t - 4 < 24) ? upfilt[t - 4] : 0.f;
        fupair[i] = p;
    }
    for (int i = lane; i < FD_LEN; i += 32) {
        const int t = i - FD_OFF;
        v2f p;
        p.x = (t     >= 0 && t     < 12) ? dnfilt[t]     : 0.f;
        p.y = (t - 1 >= 0 && t - 1 < 12) ? dnfilt[t - 1] : 0.f;
        fdpair[i] = p;
    }
    for (int i = lane; i < 32*20; i += 32) ((float*)xp)[i] = 0.f;

    // ---- load 17x17 biased input patch (zero outside image, matching conv zero-pad) ----
    const float bv = bias[plane & 255];
    const float* xpl = x + (size_t)plane * IN_HW * IN_HW;
    __builtin_prefetch(xpl + (I0y < 0 ? 0 : I0y) * IN_HW, 0, 0);   // global_prefetch_b8
    for (int i = lane; i < PATCH*PATCH; i += 32) {
        const int r = i / PATCH, c = i - r*PATCH;
        const int gi = I0y + r, gj = I0x + c;
        float v = 0.f;
        if (gi >= 0 && gi < IN_HW && gj >= 0 && gj < IN_HW)
            v = xpl[gi*IN_HW + gj] + bv;
        xp[r][c] = v;
    }
    __syncthreads();

    // ---- stage 1: up-x.  u[a][j] = sum_k xp[a][k] * fu[basex + j - 4k]
    // Rows a>=17 of xp are zero => u rows 17..31 compute to exact zeros; store all unguarded.
    #pragma unroll
    for (int mb = 0; mb < 2; ++mb) {
        #pragma unroll
        for (int nb = 0; nb < 3; ++nb) {
            v8f acc = {};
            const int tb = FU_OFF + basex + 16*nb + lo;   // lane-constant tap base
            #pragma unroll
            for (int kc = 0; kc < 5; ++kc) {
                const int gk = 4*kc + 2*hi;
                v2f a = *(const v2f*)&xp[16*mb + lo][gk]; // ds_load_b64
                v2f b = fupair[tb - 4*gk];                // ds_load_b64, pre-ordered pair
                acc = wmma4(a, b, acc);
            }
            float* dst = &ubt[16*nb + lo][16*mb + 8*hi];  // 16B aligned
            *(float4*)(dst + 0) = make_float4(acc[0], acc[1], acc[2], acc[3]); // ds_store_b128
            *(float4*)(dst + 4) = make_float4(acc[4], acc[5], acc[6], acc[7]);
        }
    }
    __syncthreads();

    // ---- stage 2: up-y + lrelu*sqrt2.  z[p][j] = act( sum_k fu[basey + p - 4k] * u[k][j] )
    // Rows p>=42 are finite don't-cares (consumed only against zero taps); store all unguarded.
    #pragma unroll
    for (int mb = 0; mb < 3; ++mb) {
        #pragma unroll
        for (int nb = 0; nb < 3; ++nb) {
            v8f acc = {};
            const int tb = FU_OFF + basey + 16*mb + lo;
            #pragma unroll
            for (int kc = 0; kc < 5; ++kc) {
                const int gk = 4*kc + 2*hi;
                v2f a = fupair[tb - 4*gk];
                v2f b = *(const v2f*)&ubt[16*nb + lo][gk]; // ds_load_b64
                acc = wmma4(a, b, acc);
            }
            #pragma unroll
            for (int r = 0; r < 8; ++r) {                  // lrelu*sqrt2 == C1*v + C2*|v|
                const float v = acc[r];
                acc[r] = __builtin_fmaf(ACT_C1, v, ACT_C2 * __builtin_fabsf(v));
            }
            float* dst = &zbt[16*nb + lo][16*mb + 8*hi];
            *(float4*)(dst + 0) = make_float4(acc[0], acc[1], acc[2], acc[3]);
            *(float4*)(dst + 4) = make_float4(acc[4], acc[5], acc[6], acc[7]);
        }
    }
    __syncthreads();

    // ---- stage 3: down-x.  t[p][mm] = sum_k z[p][k] * fd[2mm + 11 - k]
    #pragma unroll
    for (int mb = 0; mb < 3; ++mb) {
        v8f acc = {};
        const int tb = FD_OFF + 2*lo + 11;
        #pragma unroll
        for (int kc = 0; kc < 11; ++kc) {
            const int gk = 4*kc + 2*hi;
            v2f a;
            a.x = zbt[gk    ][16*mb + lo];                 // 2addr-pairable (48-dword spacing)
            a.y = zbt[gk + 1][16*mb + lo];
            v2f b = fdpair[tb - gk];                       // ds_load_b64, pre-ordered pair
            acc = wmma4(a, b, acc);
        }
        float* dst = &tbt[lo][16*mb + 8*hi];
        *(float4*)(dst + 0) = make_float4(acc[0], acc[1], acc[2], acc[3]);
        *(float4*)(dst + 4) = make_float4(acc[4], acc[5], acc[6], acc[7]);
    }
    __syncthreads();

    // ---- stage 4: down-y + store (margin crop folded into m0 = q0 + 10).
    //      out[q0y+qq][q0x+mm] = sum_k fd[2qq + 11 - k] * t[k][mm]
    {
        v8f acc = {};
        const int tb = FD_OFF + 2*lo + 11;
        #pragma unroll
        for (int kc = 0; kc < 11; ++kc) {
            const int gk = 4*kc + 2*hi;
            v2f a = fdpair[tb - gk];
            v2f b = *(const v2f*)&tbt[lo][gk];             // ds_load_b64
            acc = wmma4(a, b, acc);
        }
        float* opl = out + (size_t)plane * OUT_HW * OUT_HW;
        const int qb = q0y + 8*hi;
        const int cc = q0x + lo;
        if (q0y + TILE <= OUT_HW && q0x + TILE <= OUT_HW) {
            // interior tile (196 of 225): block-uniform branch, no per-lane guards
            #pragma unroll
            for (int r = 0; r < 8; ++r)
                opl[(qb + r) * OUT_HW + cc] = acc[r];
        } else {
            #pragma unroll
            for (int r = 0; r < 8; ++r)
                if (qb + r < OUT_HW && cc < OUT_HW)
                    opl[(qb + r) * OUT_HW + cc] = acc[r];
        }
    }
}

extern "C" void kernel_launch(void* const* d_in, const int* in_sizes, int n_in,
                              void* d_out, int out_size, void* d_ws, size_t ws_size,
                              hipStream_t stream)
{
    const float* x    = (const float*)d_in[0];   // (4,256,128,128) f32
    const float* bias = (const float*)d_in[1];   // (256,) f32
    const float* upf  = (const float*)d_in[2];   // (24,) f32
    const float* dnf  = (const float*)d_in[3];   // (12,) f32
    float* out = (float*)d_out;                  // (4,256,236,236) f32

    dim3 grid(NTILE * NTILE, 4 * 256);
    dim3 block(32);
    hipLaunchKernelGGL(afa_fused_kernel, grid, block, 0, stream, x, bias, upf, dnf, out);
}